// PointNet2Decoder_35880156791484
// MI455X (gfx1250) — compile-verified
//
#include <hip/hip_runtime.h>
#include <hip/hip_bf16.h>
#include <math.h>

typedef float v2f __attribute__((ext_vector_type(2)));
typedef float v8f __attribute__((ext_vector_type(8)));

#define WAVES_PER_BLOCK 8

// Packed activation layout ("K4"): element (c, p) of the [C,P] matrix lives at
//   Xp[ ((c>>2)*P + p)*4 + (c&3) ]
// so a WMMA B-operand lane (column p, K-pair k..k+1, k even) is one aligned b64.

// ---------------------------------------------------------------------------
// Kernel 1: 3-NN inverse-distance interpolation + concat(points1, interp),
// written directly in K4-packed layout. pts2 uses runtime strides (raw input
// [B,C,S] or computed plain [C, B*S]).
// ---------------------------------------------------------------------------
__global__ __launch_bounds__(128)
void fp_build_xcat(const float* __restrict__ xyz1, const float* __restrict__ xyz2,
                   const float* __restrict__ pts1, const float* __restrict__ pts2,
                   float* __restrict__ Xp,
                   int B, int N, int S, int D1, int C2,
                   int p2_sb, int p2_sc)
{
    int p = blockIdx.x * 128 + threadIdx.x;
    int P = B * N;
    if (p >= P) return;
    int b = p / N;
    int n = p - b * N;

    const float* x1b = xyz1 + (size_t)b * 3 * N;
    float x = x1b[n], y = x1b[N + n], z = x1b[2 * N + n];

    const float* x2b = xyz2 + (size_t)b * 3 * S;
    float d0 = 3.0e38f, d1 = 3.0e38f, d2 = 3.0e38f;
    int i0 = 0, i1 = 0, i2 = 0;
    for (int s = 0; s < S; ++s) {
        float dx = x - x2b[s];
        float dy = y - x2b[S + s];
        float dz = z - x2b[2 * S + s];
        float d = dx * dx + dy * dy + dz * dz;
        if (d < d0)      { d2 = d1; i2 = i1; d1 = d0; i1 = i0; d0 = d; i0 = s; }
        else if (d < d1) { d2 = d1; i2 = i1; d1 = d;  i1 = s; }
        else if (d < d2) { d2 = d;  i2 = s; }
    }
    float w0 = 1.0f / (d0 + 1e-8f);
    float w1 = 1.0f / (d1 + 1e-8f);
    float w2 = 1.0f / (d2 + 1e-8f);
    float wsum = w0 + w1 + w2;
    w0 /= wsum; w1 /= wsum; w2 /= wsum;

    // points1 passthrough -> rows [0, D1)
    for (int c = 0; c < D1; ++c)
        Xp[((size_t)(c >> 2) * P + p) * 4 + (c & 3)] =
            pts1[((size_t)b * D1 + c) * N + n];

    // interpolated points2 -> rows [D1, D1+C2)
    const float* p2b = pts2 + (size_t)b * p2_sb;
    for (int c2 = 0; c2 < C2; ++c2) {
        const float* pc = p2b + (size_t)c2 * p2_sc;
        int c = D1 + c2;
        Xp[((size_t)(c >> 2) * P + p) * 4 + (c & 3)] =
            w0 * pc[i0] + w1 * pc[i1] + w2 * pc[i2];
    }
}

// ---------------------------------------------------------------------------
// Kernel 2: fp32 WMMA GEMM  Y[O,P] = W[O,C] * X[C,P] + bias
// One wave per 16x64 output block (4 accumulators), K-step 4 via
// V_WMMA_F32_16X16X4_F32. O is always a multiple of 16 (classifier weights
// are zero-padded), so the K-loop is divergence-free: 1 b64 A-load + 4 b64
// B-loads + 4 WMMAs per step.
// outMode 0: Y row-major [O,P].  outMode 1: scatter to [B, Oreal, N].
// ---------------------------------------------------------------------------
__global__ __launch_bounds__(WAVES_PER_BLOCK * 32)
void gemm_wmma_f32(const float* __restrict__ W, const float* __restrict__ bias,
                   const float* __restrict__ Xp, float* __restrict__ Y,
                   int O, int C, int P, int outMode, int Oreal, int N)
{
    int tilesP = P >> 6;                                  // 64 cols per wave
    int wave = blockIdx.x * WAVES_PER_BLOCK + (threadIdx.x >> 5);
    if (wave >= (O >> 4) * tilesP) return;                // wave-uniform exit

    int to = (wave / tilesP) << 4;
    int tp = (wave % tilesP) << 6;
    int lane = threadIdx.x & 31;
    int mn   = lane & 15;
    int hi   = lane >> 4;
    int khalf = hi << 1;                                  // 0 or 2

    v8f acc[4];
#pragma unroll
    for (int r = 0; r < 8; ++r) {
        float bv = bias[to + r + (hi << 3)];
#pragma unroll
        for (int j = 0; j < 4; ++j) acc[j][r] = bv;
    }

    const float* Wrow  = W + (size_t)(to + mn) * C + khalf;
    const float* Xbase = Xp + (size_t)(tp + mn) * 4 + khalf;
    size_t xrowstride = (size_t)P * 4;                    // per K-group of 4

#pragma unroll 2
    for (int k0 = 0; k0 < C; k0 += 4) {
        v2f a = *(const v2f*)(Wrow + k0);
        const float* xb = Xbase + (size_t)(k0 >> 2) * xrowstride;
        v2f b0 = *(const v2f*)(xb);
        v2f b1 = *(const v2f*)(xb + 64);
        v2f b2 = *(const v2f*)(xb + 128);
        v2f b3 = *(const v2f*)(xb + 192);
        acc[0] = __builtin_amdgcn_wmma_f32_16x16x4_f32(false, a, false, b0, (short)0, acc[0], false, false);
        acc[1] = __builtin_amdgcn_wmma_f32_16x16x4_f32(false, a, false, b1, (short)0, acc[1], false, false);
        acc[2] = __builtin_amdgcn_wmma_f32_16x16x4_f32(false, a, false, b2, (short)0, acc[2], false, false);
        acc[3] = __builtin_amdgcn_wmma_f32_16x16x4_f32(false, a, false, b3, (short)0, acc[3], false, false);
    }

#pragma unroll
    for (int r = 0; r < 8; ++r) {
        int row = to + r + (hi << 3);
        if (row < Oreal) {
#pragma unroll
            for (int j = 0; j < 4; ++j) {
                int col = tp + (j << 4) + mn;
                float v = acc[j][r];
                if (outMode == 0) {
                    Y[(size_t)row * P + col] = v;
                } else {
                    int bb = col / N;
                    int nn = col - bb * N;
                    Y[((size_t)bb * Oreal + row) * N + nn] = v;
                }
            }
        }
    }
}

// ---------------------------------------------------------------------------
// Kernel 3: train-mode BN statistics per channel (biased var, eps = 1e-5)
// ---------------------------------------------------------------------------
__global__ __launch_bounds__(256)
void bn_stats(const float* __restrict__ Y, int P,
              float* __restrict__ mean, float* __restrict__ rstd)
{
    int o = blockIdx.x;
    const float* row = Y + (size_t)o * P;
    float s = 0.0f, s2 = 0.0f;
    for (int i = threadIdx.x; i < P; i += 256) {
        float v = row[i];
        s += v; s2 += v * v;
    }
    __shared__ float sh[256], sh2[256];
    sh[threadIdx.x] = s; sh2[threadIdx.x] = s2;
    __syncthreads();
    for (int d = 128; d > 0; d >>= 1) {
        if (threadIdx.x < d) {
            sh[threadIdx.x]  += sh[threadIdx.x + d];
            sh2[threadIdx.x] += sh2[threadIdx.x + d];
        }
        __syncthreads();
    }
    if (threadIdx.x == 0) {
        float invP = 1.0f / (float)P;
        float m = sh[0] * invP;
        float var = sh2[0] * invP - m * m;
        mean[o] = m;
        rstd[o] = rsqrtf(var + 1e-5f);
    }
}

// ---------------------------------------------------------------------------
// Kernel 4: BN apply + ReLU. mode 0: write K4-packed to dst (feeds next GEMM).
//           mode 1: write plain [O,P] in-place (feeds next stage's 3-NN gather).
// ---------------------------------------------------------------------------
__global__ __launch_bounds__(256)
void bn_relu(float* __restrict__ Y, float* __restrict__ dst,
             const float* __restrict__ mean, const float* __restrict__ rstd,
             const float* __restrict__ g, const float* __restrict__ be,
             int P, int mode)
{
    int o = blockIdx.y;
    int i = blockIdx.x * 256 + threadIdx.x;
    if (i >= P) return;
    float v = (Y[(size_t)o * P + i] - mean[o]) * rstd[o] * g[o] + be[o];
    v = fmaxf(v, 0.0f);
    if (mode == 0)
        dst[((size_t)(o >> 2) * P + i) * 4 + (o & 3)] = v;
    else
        Y[(size_t)o * P + i] = v;
}

// ---------------------------------------------------------------------------
// Kernel 5: zero-pad classifier weights 13x128 -> 16x128 (+ bias) so the
// GEMM K-loop never needs per-lane guards.
// ---------------------------------------------------------------------------
__global__ __launch_bounds__(256)
void pad_conv_out(const float* __restrict__ w, const float* __restrict__ b,
                  float* __restrict__ wpad, float* __restrict__ bpad)
{
    int i = blockIdx.x * 256 + threadIdx.x;
    if (i < 16 * 128) {
        int o = i >> 7;
        wpad[i] = (o < 13) ? w[i] : 0.0f;
    } else if (i < 16 * 128 + 16) {
        int o = i - 16 * 128;
        bpad[o] = (o < 13) ? b[o] : 0.0f;
    }
}

// ---------------------------------------------------------------------------
// Host orchestration
// ---------------------------------------------------------------------------
struct FpLayer { const float *w, *b, *g, *be; int O, C; };

extern "C" void kernel_launch(void* const* d_in, const int* in_sizes, int n_in,
                              void* d_out, int out_size, void* d_ws, size_t ws_size,
                              hipStream_t stream)
{
    (void)in_sizes; (void)n_in; (void)out_size; (void)ws_size;

    const float* l0_xyz    = (const float*)d_in[0];
    const float* l1_xyz    = (const float*)d_in[2];
    const float* l1_points = (const float*)d_in[3];
    const float* l2_xyz    = (const float*)d_in[4];
    const float* l2_points = (const float*)d_in[5];
    const float* l3_xyz    = (const float*)d_in[6];
    const float* l3_points = (const float*)d_in[7];
    const float* l4_xyz    = (const float*)d_in[8];
    const float* l4_points = (const float*)d_in[9];

    int pi = 10;
    auto nextLayer = [&](int O, int C) -> FpLayer {
        FpLayer L{(const float*)d_in[pi], (const float*)d_in[pi + 1],
                  (const float*)d_in[pi + 2], (const float*)d_in[pi + 3], O, C};
        pi += 4;
        return L;
    };
    FpLayer fp4L[2] = { nextLayer(256, 768), nextLayer(256, 256) };
    FpLayer fp3L[2] = { nextLayer(256, 384), nextLayer(256, 256) };
    FpLayer fp2L[2] = { nextLayer(256, 320), nextLayer(128, 256) };
    FpLayer fp1L[3] = { nextLayer(128, 128), nextLayer(128, 128), nextLayer(128, 128) };
    FpLayer conv1   = nextLayer(128, 128);
    const float* conv_out_w = (const float*)d_in[pi];
    const float* conv_out_b = (const float*)d_in[pi + 1];

    const int B = 16;
    const int NUM_CLASS = 13;

    // Workspace (floats): big0 = K4-packed activations, big1 = plain GEMM out,
    // prev = previous-stage points (plain [C, B*S]), stats, padded classifier.
    float* ws = (float*)d_ws;
    const size_t BIG = (size_t)128 * 65536;           // 8,388,608 floats each
    float* big0 = ws;
    float* big1 = ws + BIG;
    float* prev = ws + 2 * BIG;                       // up to 128*16384 floats
    float* mean = prev + (size_t)128 * 16384;
    float* rstd = mean + 256;
    float* wpad = rstd + 256;                         // 16*128
    float* bpad = wpad + 16 * 128;                    // 16

    pad_conv_out<<<dim3((16 * 128 + 16 + 255) / 256), dim3(256), 0, stream>>>(
        conv_out_w, conv_out_b, wpad, bpad);

    auto gemm = [&](const float* W, const float* bias, const float* Xin, float* Yout,
                    int O, int C, int P, int outMode, int Oreal, int Ncols) {
        int tiles = (O >> 4) * (P >> 6);
        int blocks = (tiles + WAVES_PER_BLOCK - 1) / WAVES_PER_BLOCK;
        gemm_wmma_f32<<<dim3(blocks), dim3(WAVES_PER_BLOCK * 32), 0, stream>>>(
            W, bias, Xin, Yout, O, C, P, outMode, Oreal, Ncols);
    };
    auto bn = [&](float* Yio, float* dst, const float* g, const float* be,
                  int O, int P, int mode) {
        bn_stats<<<dim3(O), dim3(256), 0, stream>>>(Yio, P, mean, rstd);
        bn_relu<<<dim3((P + 255) / 256, O), dim3(256), 0, stream>>>(
            Yio, dst, mean, rstd, g, be, P, mode);
    };
    // Runs one FP stage. big0 holds packed activations, big1 plain GEMM output.
    // If finalPlain, the stage result is left plain in big1 (for 3-NN gather);
    // otherwise it ends K4-packed in big0 (for the next GEMM).
    auto run_stage = [&](const float* xyz1, const float* xyz2,
                         const float* pts1, int D1,
                         const float* pts2, int C2, int p2_sb, int p2_sc,
                         int N, int S, const FpLayer* Ls, int nL, bool finalPlain) {
        int P = B * N;
        fp_build_xcat<<<dim3((P + 127) / 128), dim3(128), 0, stream>>>(
            xyz1, xyz2, pts1, pts2, big0, B, N, S, D1, C2, p2_sb, p2_sc);
        for (int i = 0; i < nL; ++i) {
            gemm(Ls[i].w, Ls[i].b, big0, big1, Ls[i].O, Ls[i].C, P, 0, Ls[i].O, 0);
            bool plain = finalPlain && (i == nL - 1);
            bn(big1, big0, Ls[i].g, Ls[i].be, Ls[i].O, P, plain ? 1 : 0);
        }
    };

    // fp4: N=64, S=16, points1 = l3_points [B,256,64], points2 = l4_points [B,512,16]
    run_stage(l3_xyz, l4_xyz, l3_points, 256,
              l4_points, 512, /*sb=*/512 * 16, /*sc=*/16, 64, 16, fp4L, 2, true);
    hipMemcpyAsync(prev, big1, sizeof(float) * 256 * 1024, hipMemcpyDeviceToDevice, stream);

    // fp3: N=256, S=64, points2 = l3p plain [C=256, B*S=1024] -> sb=S, sc=B*S
    run_stage(l2_xyz, l3_xyz, l2_points, 128,
              prev, 256, /*sb=*/64, /*sc=*/1024, 256, 64, fp3L, 2, true);
    hipMemcpyAsync(prev, big1, sizeof(float) * 256 * 4096, hipMemcpyDeviceToDevice, stream);

    // fp2: N=1024, S=256, points2 = l2p plain [C=256, B*S=4096]
    run_stage(l1_xyz, l2_xyz, l1_points, 64,
              prev, 256, /*sb=*/256, /*sc=*/4096, 1024, 256, fp2L, 2, true);
    hipMemcpyAsync(prev, big1, sizeof(float) * 128 * 16384, hipMemcpyDeviceToDevice, stream);

    // fp1: N=4096, S=1024, points1 = None, points2 = l1p plain [C=128, B*S=16384]
    // Result stays K4-packed in big0 (feeds conv1 GEMM).
    run_stage(l0_xyz, l1_xyz, nullptr, 0,
              prev, 128, /*sb=*/1024, /*sc=*/16384, 4096, 1024, fp1L, 3, false);

    // head: conv1 + BN + ReLU (packed), then conv_out scattered to [B,13,4096]
    int P0 = B * 4096;
    gemm(conv1.w, conv1.b, big0, big1, 128, 128, P0, 0, 128, 0);
    bn(big1, big0, conv1.g, conv1.be, 128, P0, 0);
    gemm(wpad, bpad, big0, (float*)d_out, 16, 128, P0, 1, NUM_CLASS, 4096);
}